// AttentionMechanism_34832184770732
// MI455X (gfx1250) — compile-verified
//
#include <hip/hip_runtime.h>

#define D_CONST 64
#define HEADS   4
#define NH      256          // HEADS * D
#define N_NODES 50000
#define TILE_E  128          // edges per block (8 M-tiles of 16)
#define SA_STRIDE 68         // padded LDS row stride (floats) -> conflict-free

typedef __attribute__((ext_vector_type(2))) float v2f;
typedef __attribute__((ext_vector_type(8))) float v8f;

__device__ __forceinline__ void async_copy16(unsigned lds_byte_off, const float* g) {
    // global -> LDS async copy, 16 bytes per lane (ASYNCcnt tracked)
    asm volatile("global_load_async_to_lds_b128 %0, %1, off"
                 :: "v"(lds_byte_off), "v"(g) : "memory");
}

__device__ __forceinline__ void wait_async0() {
#if __has_builtin(__builtin_amdgcn_s_wait_asynccnt)
    __builtin_amdgcn_s_wait_asynccnt(0);
#else
    asm volatile("s_wait_asynccnt 0" ::: "memory");
#endif
}

// ---------------------------------------------------------------------------
// Pass 0: init node max (-inf) and node sum (0) buffers (L2-resident, 51MB ea)
// ---------------------------------------------------------------------------
__global__ __launch_bounds__(256) void gat_init_nodes(float4* __restrict__ node_max,
                                                      float4* __restrict__ node_sum,
                                                      int n4) {
    int i = blockIdx.x * blockDim.x + threadIdx.x;
    if (i < n4) {
        const float ninf = -__builtin_inff();
        node_max[i] = make_float4(ninf, ninf, ninf, ninf);
        node_sum[i] = make_float4(0.f, 0.f, 0.f, 0.f);
    }
}

// ---------------------------------------------------------------------------
// Pass 1: logits = messages @ W via V_WMMA_F32_16X16X4_F32.
//  Block = 256 threads (8 waves) covers 128 edges (8 M-tiles).
//  A tile (128x64 f32) staged to LDS with global_load_async_to_lds_b128,
//  padded stride -> conflict-free ds_load_2addr_b64 A-fragment reads.
//  Wave w owns N-tiles [32w,32w+16) and [32w+16,32w+32); its 16 k-step
//  B-fragments are hoisted into registers and reused across all 8 M-tiles.
//  Per M-tile, all 16 A-fragments are batch-loaded (one s_wait_dscnt),
//  then 32 WMMAs issue back-to-back.
// ---------------------------------------------------------------------------
__global__ __launch_bounds__(256) void gat_gemm_max(const float* __restrict__ msgs,
                                                    const long long* __restrict__ src,
                                                    const float* __restrict__ W,
                                                    float* __restrict__ logits,
                                                    float* __restrict__ node_max,
                                                    int E) {
    __shared__ float sA[TILE_E * SA_STRIDE];      // 128*68*4 = 34.8 KB

    const int base_e = blockIdx.x * TILE_E;
    const int lane = threadIdx.x & 31;
    const int wave = threadIdx.x >> 5;            // 0..7
    const int hi   = lane >> 4;                   // 0: K+0/K+1 lanes, 1: K+2/K+3
    const int l16  = lane & 15;

    // ---- stage A tile: 128 rows x 64 floats = 2048 x 16B chunks, 8/thread ----
#pragma unroll
    for (int i = 0; i < 8; ++i) {
        const int chunk = threadIdx.x + i * 256;  // 0..2047
        const int row   = chunk >> 4;             // 0..127
        const int seg   = chunk & 15;             // 16B segment in row
        int e = base_e + row;
        if (e >= E) e = E - 1;                    // clamp (no EXEC divergence)
        const float* g = msgs + (size_t)e * D_CONST + seg * 4;
        const unsigned loff =
            (unsigned)(uintptr_t)(&sA[row * SA_STRIDE + seg * 4]);
        async_copy16(loff, g);
    }

    // ---- hoist B fragments: 16 k-steps x 2 N-tiles, reused for 8 M-tiles ----
    const int n0 = wave * 32;
    const int n1 = wave * 32 + 16;
    v2f B0[16], B1[16];
#pragma unroll
    for (int kk = 0; kk < 16; ++kk) {
        const int ka = kk * 4 + hi * 2;
        B0[kk].x = W[(size_t)ka * NH + n0 + l16];
        B0[kk].y = W[(size_t)(ka + 1) * NH + n0 + l16];
        B1[kk].x = W[(size_t)ka * NH + n1 + l16];
        B1[kk].y = W[(size_t)(ka + 1) * NH + n1 + l16];
    }

    wait_async0();
    __syncthreads();

    // ---- 8 M-tiles: batch A frags from LDS, B frags from registers ----
#pragma unroll 1
    for (int m = 0; m < 8; ++m) {
        // Batch-load all 16 A fragments for this M-tile (clause of
        // ds_load_2addr_b64, single s_wait_dscnt before the WMMA burst).
        v2f A[16];
#pragma unroll
        for (int kk = 0; kk < 16; ++kk) {
            const int ka = kk * 4 + hi * 2;
            A[kk] = *(const v2f*)&sA[(m * 16 + l16) * SA_STRIDE + ka];
        }

        v8f c0 = {};
        v8f c1 = {};
#pragma unroll
        for (int kk = 0; kk < 16; ++kk) {
            c0 = __builtin_amdgcn_wmma_f32_16x16x4_f32(false, A[kk], false, B0[kk],
                                                       (short)0, c0, false, false);
            c1 = __builtin_amdgcn_wmma_f32_16x16x4_f32(false, A[kk], false, B1[kk],
                                                       (short)0, c1, false, false);
        }

        // C/D: VGPR r, lanes 0-15 -> M=r, lanes 16-31 -> M=r+8; N = lane%16
#pragma unroll
        for (int r = 0; r < 8; ++r) {
            const int e = base_e + m * 16 + r + hi * 8;
            if (e < E) {
                const long long s = src[e];
                const size_t obase = (size_t)e * NH;
                const size_t nbase = (size_t)s * NH;
                const float v0 = c0[r];
                const float v1 = c1[r];
                logits[obase + n0 + l16] = v0;
                logits[obase + n1 + l16] = v1;
                __hip_atomic_fetch_max(&node_max[nbase + n0 + l16], v0,
                                       __ATOMIC_RELAXED, __HIP_MEMORY_SCOPE_AGENT);
                __hip_atomic_fetch_max(&node_max[nbase + n1 + l16], v1,
                                       __ATOMIC_RELAXED, __HIP_MEMORY_SCOPE_AGENT);
            }
        }
    }
}

// ---------------------------------------------------------------------------
// Pass 2: w = exp(logit - mx[src]); overwrite buf; atomicAdd into node_sum.
// float4 vectorized; node buffers are L2-resident gathers.
// ---------------------------------------------------------------------------
__global__ __launch_bounds__(256) void gat_exp_sum(float4* __restrict__ buf,
                                                   const long long* __restrict__ src,
                                                   const float4* __restrict__ node_max,
                                                   float* __restrict__ node_sum,
                                                   size_t total4) {
    size_t i = (size_t)blockIdx.x * blockDim.x + threadIdx.x;
    if (i >= total4) return;
    const size_t e  = i >> 6;                 // 64 float4 groups per edge
    const int    c4 = (int)(i & 63);
    const long long s = src[e];
    const float4 l  = buf[i];
    const float4 mx = node_max[(size_t)s * 64 + c4];
    float4 w;
    w.x = __expf(l.x - mx.x);
    w.y = __expf(l.y - mx.y);
    w.z = __expf(l.z - mx.z);
    w.w = __expf(l.w - mx.w);
    buf[i] = w;
    float* ns = node_sum + (size_t)s * NH + c4 * 4;
    atomicAdd(ns + 0, w.x);
    atomicAdd(ns + 1, w.y);
    atomicAdd(ns + 2, w.z);
    atomicAdd(ns + 3, w.w);
}

// ---------------------------------------------------------------------------
// Pass 3: probs = w / s[src]   (float4 vectorized)
// ---------------------------------------------------------------------------
__global__ __launch_bounds__(256) void gat_normalize(float4* __restrict__ buf,
                                                     const long long* __restrict__ src,
                                                     const float4* __restrict__ node_sum,
                                                     size_t total4) {
    size_t i = (size_t)blockIdx.x * blockDim.x + threadIdx.x;
    if (i >= total4) return;
    const size_t e  = i >> 6;
    const int    c4 = (int)(i & 63);
    const long long s = src[e];
    const float4 w  = buf[i];
    const float4 sm = node_sum[(size_t)s * 64 + c4];
    float4 p;
    p.x = w.x / sm.x;
    p.y = w.y / sm.y;
    p.z = w.z / sm.z;
    p.w = w.w / sm.w;
    buf[i] = p;
}

// ---------------------------------------------------------------------------
extern "C" void kernel_launch(void* const* d_in, const int* in_sizes, int n_in,
                              void* d_out, int out_size, void* d_ws, size_t ws_size,
                              hipStream_t stream) {
    const float*     msgs = (const float*)d_in[0];
    const long long* eidx = (const long long*)d_in[1];  // [2, E]; row 0 = src
    const float*     W    = (const float*)d_in[2];

    float* out = (float*)d_out;                  // logits -> w -> probs
    float* node_max = (float*)d_ws;              // [N_NODES * NH]
    float* node_sum = node_max + (size_t)N_NODES * NH;

    const int E = in_sizes[0] / D_CONST;
    const long long* src = eidx;                 // first E entries

    const int n4 = N_NODES * NH / 4;
    gat_init_nodes<<<(n4 + 255) / 256, 256, 0, stream>>>(
        (float4*)node_max, (float4*)node_sum, n4);

    gat_gemm_max<<<(E + TILE_E - 1) / TILE_E, 256, 0, stream>>>(
        msgs, src, W, out, node_max, E);

    const size_t total4 = (size_t)E * NH / 4;
    const int eb = (int)((total4 + 255) / 256);
    gat_exp_sum<<<eb, 256, 0, stream>>>((float4*)out, src, (float4*)node_max,
                                        node_sum, total4);
    gat_normalize<<<eb, 256, 0, stream>>>((float4*)out, src, (float4*)node_sum,
                                          total4);
}